// EdgesMultiheadAttention_35485019799988
// MI455X (gfx1250) — compile-verified
//
#include <hip/hip_runtime.h>
#include <math.h>

// EdgesMultiheadAttention for MI455X (gfx1250, wave32).
// Matmuls use V_WMMA_F32_16X16X4_F32 (fp32-exact matrix path).
// Tiles are staged Global->LDS with the CDNA5 async copy engine
// (GLOBAL_LOAD_ASYNC_TO_LDS_B128, ASYNCcnt) and consumed via DS loads.
#define B_  4
#define T_  1024
#define S_  1024
#define D_  512
#define H_  8
#define DH_ 64
#define BIAS_DIM_ 64

typedef float v2f __attribute__((ext_vector_type(2)));
typedef float v8f __attribute__((ext_vector_type(8)));

// ------------------------------------------------ CDNA5 async-copy helpers

// Low 32 bits of a generic pointer to a __shared__ object = LDS byte address.
__device__ __forceinline__ unsigned lds_addr(const void* p) {
    return (unsigned)(unsigned long long)p;
}

// Per-lane 16B Global -> LDS async copy (tracked with ASYNCcnt).
__device__ __forceinline__ void async_copy16(unsigned lds_dst, const float* gsrc) {
    asm volatile("global_load_async_to_lds_b128 %0, %1, off"
                 :: "v"(lds_dst), "v"(gsrc)
                 : "memory");
}

__device__ __forceinline__ void wait_async0() {
#if __has_builtin(__builtin_amdgcn_s_wait_asynccnt)
    __builtin_amdgcn_s_wait_asynccnt(0);
#else
    asm volatile("s_wait_asynccnt 0" ::: "memory");
#endif
}

// ---------------------------------------------------------------- utilities

__global__ void zero_f32_kernel(float* __restrict__ p, int n) {
    int i = blockIdx.x * blockDim.x + threadIdx.x;
    if (i < n) p[i] = 0.0f;
}

// proj[j] = bias_embs[j,:] . bias_scalar[:,0]   (64 x 64 -> 64)
__global__ void bias_proj_kernel(const float* __restrict__ bias_embs,
                                 const float* __restrict__ bias_scalar,
                                 float* __restrict__ proj) {
    int j = threadIdx.x;
    if (j < BIAS_DIM_) {
        float acc = 0.0f;
        #pragma unroll
        for (int d = 0; d < DH_; ++d) acc += bias_embs[j * DH_ + d] * bias_scalar[d];
        proj[j] = acc;
    }
}

// bias_flat[edges0[e]] += proj[edges1[e]]
__global__ void edge_scatter_kernel(const int* __restrict__ edges0,
                                    const int* __restrict__ edges1,
                                    const float* __restrict__ proj,
                                    float* __restrict__ bias_flat, int E) {
    int e = blockIdx.x * blockDim.x + threadIdx.x;
    if (e < E) atomicAdd(&bias_flat[edges0[e]], proj[edges1[e]]);
}

// summed[b,s,h] = sum_d key[b,s,h,d]
__global__ void summed_keys_kernel(const float* __restrict__ key,
                                   float* __restrict__ summed) {
    int i = blockIdx.x * blockDim.x + threadIdx.x;   // over B*S*H
    if (i >= B_ * S_ * H_) return;
    int h = i % H_;
    int s = (i / H_) % S_;
    int b = i / (H_ * S_);
    const float* kr = key + ((size_t)b * S_ + s) * D_ + h * DH_;
    float acc = 0.0f;
    #pragma unroll 8
    for (int d = 0; d < DH_; ++d) acc += kr[d];
    summed[i] = acc;
}

// -------------------------------------------------- scores: Q K^T via WMMA
// 128 threads = 4 waves per block; block computes a [16 t x 64 s] score tile.
// Q tile (16x64) and K tile (64x64) staged into LDS with async b128 copies.
// f32 WMMA fragment layout (ISA 7.12.2):
//   A 16x4:  lanes 0-15 row M=lane hold K={0,1}; lanes 16-31 same rows, K={2,3}
//   B 4x16:  mirrored: lanes 0-15 col N=lane hold K={0,1}; upper lanes K={2,3}
//   C/D 16x16: vgpr j -> (M=j + 8*(lane>=16), N=lane&15)
#define QSTR 68   // padded LDS row stride (floats): conflict-free column reads
#define KSTR 68

__global__ __launch_bounds__(128)
void scores_wmma_kernel(const float* __restrict__ query,
                        const float* __restrict__ key,
                        const unsigned char* __restrict__ mask, // [B,S] bool
                        const float* __restrict__ bias_flat,    // [B,T,S]
                        const float* __restrict__ summed,       // [B,S,H]
                        float* __restrict__ alpha) {            // [B,H,T,S]
    __shared__ float qs[16 * QSTR];
    __shared__ float ks[64 * KSTR];

    const int tid  = threadIdx.x;
    const int lane = tid & 31;
    const int wave = tid >> 5;
    const int s0 = blockIdx.x * 64;
    const int t0 = blockIdx.y * 16;
    const int bh = blockIdx.z;
    const int b = bh / H_, h = bh % H_;

    const float* qbase = query + ((size_t)b * T_ + t0) * D_ + h * DH_;
    const float* kbase = key   + ((size_t)b * S_ + s0) * D_ + h * DH_;

    // Stage Q: 16 rows x 16 chunks of 4 floats  (2 async copies / lane)
    #pragma unroll
    for (int c = tid; c < 256; c += 128) {
        const int r = c >> 4, c4 = (c & 15) * 4;
        async_copy16(lds_addr(&qs[r * QSTR + c4]), qbase + (size_t)r * D_ + c4);
    }
    // Stage K: 64 rows x 16 chunks             (8 async copies / lane)
    #pragma unroll
    for (int c = tid; c < 1024; c += 128) {
        const int r = c >> 4, c4 = (c & 15) * 4;
        async_copy16(lds_addr(&ks[r * KSTR + c4]), kbase + (size_t)r * D_ + c4);
    }
    wait_async0();
    __syncthreads();

    const int row   = lane & 15;
    const int khalf = lane >> 4;
    v8f c = {};
    #pragma unroll
    for (int i = 0; i < DH_ / 4; ++i) {            // 16 WMMA steps over dh=64
        const int kb = i * 4 + khalf * 2;
        v2f a;  a.x  = qs[row * QSTR + kb];
                a.y  = qs[row * QSTR + kb + 1];
        v2f bb; bb.x = ks[(wave * 16 + row) * KSTR + kb];
                bb.y = ks[(wave * 16 + row) * KSTR + kb + 1];
        c = __builtin_amdgcn_wmma_f32_16x16x4_f32(false, a, false, bb,
                                                  (short)0, c, false, false);
    }

    // epilogue: + bias[b,t,s] * summed_keys[b,s,h], * 1/sqrt(dh), mask
    const int s = s0 + wave * 16 + row;                        // N index
    const float skv = summed[((size_t)b * S_ + s) * H_ + h];
    const bool msk = mask[(size_t)b * S_ + s] != 0;
    #pragma unroll
    for (int j = 0; j < 8; ++j) {
        const int t = t0 + j + khalf * 8;                      // M index
        const float bias = bias_flat[((size_t)b * T_ + t) * S_ + s];
        float v = (c[j] + bias * skv) * 0.125f;                // 1/sqrt(64)
        if (msk) v = -1e9f;
        alpha[((size_t)bh * T_ + t) * S_ + s] = v;
    }
}

// --------------------------------------------------------- rowwise softmax
// One 256-thread block per (b,h,t) row of S=1024. exp held in registers so
// alpha is read once / written once.
__global__ void softmax_rows_kernel(float* __restrict__ alpha) {
    const int row = blockIdx.x;                 // B*H*T rows
    float* p = alpha + (size_t)row * S_;
    const int tid = threadIdx.x;                // 0..255
    __shared__ float redm[8];
    __shared__ float reds[8];

    float x[4];
    #pragma unroll
    for (int j = 0; j < 4; ++j) x[j] = p[tid + j * 256];

    float m = fmaxf(fmaxf(x[0], x[1]), fmaxf(x[2], x[3]));
    #pragma unroll
    for (int off = 16; off > 0; off >>= 1) m = fmaxf(m, __shfl_xor(m, off, 32));
    if ((tid & 31) == 0) redm[tid >> 5] = m;
    __syncthreads();
    float rowmax = redm[0];
    #pragma unroll
    for (int w = 1; w < 8; ++w) rowmax = fmaxf(rowmax, redm[w]);

    float e[4];
    float lsum = 0.0f;
    #pragma unroll
    for (int j = 0; j < 4; ++j) { e[j] = expf(x[j] - rowmax); lsum += e[j]; }
    #pragma unroll
    for (int off = 16; off > 0; off >>= 1) lsum += __shfl_xor(lsum, off, 32);
    if ((tid & 31) == 0) reds[tid >> 5] = lsum;
    __syncthreads();
    float rowsum = reds[0];
    #pragma unroll
    for (int w = 1; w < 8; ++w) rowsum += reds[w];

    const float inv = 1.0f / rowsum;
    #pragma unroll
    for (int j = 0; j < 4; ++j) p[tid + j * 256] = e[j] * inv;
}

// ------------------------------------------------- out = alpha @ V via WMMA
// 128 threads = 4 waves; block computes a [16 t x 64 d] out tile.
// Loop over S in 64-wide chunks; V (64x64) and alpha (16x64) tiles staged
// into double-buffered LDS with async copies so the ASYNC engine overlaps
// the WMMA chain. A-fragments are shared by all 4 waves from LDS.
#define VSTR 64
#define ASTR 68

__global__ __launch_bounds__(128)
void pv_wmma_kernel(const float* __restrict__ value,   // [B,S,D]
                    const float* __restrict__ alpha,   // [B,H,T,S]
                    float* __restrict__ out) {         // [B,T,D]
    __shared__ float vsb[2][64 * VSTR];   // 2 x 16 KB
    __shared__ float asb[2][16 * ASTR];   // 2 x 4.25 KB

    const int tid  = threadIdx.x;
    const int lane = tid & 31;
    const int wave = tid >> 5;
    const int t0 = blockIdx.x * 16;
    const int bh = blockIdx.y;
    const int b = bh / H_, h = bh % H_;
    const int d0 = wave * 16;             // this wave's 16 output columns

    const float* vbase = value + (size_t)b * S_ * D_ + h * DH_;
    const float* abase = alpha + ((size_t)bh * T_ + t0) * S_;

    const int row   = lane & 15;
    const int khalf = lane >> 4;

    // ---- stage chunk `ck` into buffer `buf`
    auto stage = [&](int ck, int buf) {
        const int sbeg = ck * 64;
        // V chunk: 64 rows x 16 chunks of 4 floats (8 copies / lane)
        #pragma unroll
        for (int c = tid; c < 1024; c += 128) {
            const int r = c >> 4, c4 = (c & 15) * 4;
            async_copy16(lds_addr(&vsb[buf][r * VSTR + c4]),
                         vbase + (size_t)(sbeg + r) * D_ + c4);
        }
        // alpha chunk: 16 rows x 16 chunks (2 copies / lane)
        #pragma unroll
        for (int c = tid; c < 256; c += 128) {
            const int r = c >> 4, c4 = (c & 15) * 4;
            async_copy16(lds_addr(&asb[buf][r * ASTR + c4]),
                         abase + (size_t)r * S_ + sbeg + c4);
        }
    };

    stage(0, 0);
    v8f acc = {};
    for (int ck = 0; ck < S_ / 64; ++ck) {
        const int buf = ck & 1;
        wait_async0();          // my copies for chunk ck have landed
        __syncthreads();        // everyone's copies landed; prev compute done
        if (ck + 1 < S_ / 64) stage(ck + 1, buf ^ 1);
        #pragma unroll
        for (int i = 0; i < 16; ++i) {   // 16 WMMA steps over 64 K-values
            const int kb = i * 4 + khalf * 2;
            v2f a;  a.x  = asb[buf][row * ASTR + kb];
                    a.y  = asb[buf][row * ASTR + kb + 1];
            v2f bb; bb.x = vsb[buf][kb * VSTR + d0 + row];
                    bb.y = vsb[buf][(kb + 1) * VSTR + d0 + row];
            acc = __builtin_amdgcn_wmma_f32_16x16x4_f32(false, a, false, bb,
                                                        (short)0, acc, false, false);
        }
    }

    #pragma unroll
    for (int j = 0; j < 8; ++j) {
        const int t = t0 + j + khalf * 8;
        out[((size_t)b * T_ + t) * D_ + h * DH_ + d0 + row] = acc[j];
    }
}

// ------------------------------------------------------------------ launch
extern "C" void kernel_launch(void* const* d_in, const int* in_sizes, int n_in,
                              void* d_out, int out_size, void* d_ws, size_t ws_size,
                              hipStream_t stream) {
    const float*         query       = (const float*)d_in[0];
    const float*         key         = (const float*)d_in[1];
    const float*         value       = (const float*)d_in[2];
    const unsigned char* mask        = (const unsigned char*)d_in[3]; // bool [B,1,1,S]
    const int*           edges0      = (const int*)d_in[4];
    const int*           edges1      = (const int*)d_in[5];
    const float*         bias_embs   = (const float*)d_in[6];
    const float*         bias_scalar = (const float*)d_in[7];
    const int E = in_sizes[4];

    float* out_f = (float*)d_out;                       // [B,T,D]
    float* alpha = out_f + (size_t)B_ * T_ * D_;        // [B,H,T,S]

    // workspace: bias_flat (16 MB) | summed_keys (128 KB) | proj (256 B)
    float* bias_flat = (float*)d_ws;                    // B*T*S
    float* summed    = bias_flat + (size_t)B_ * T_ * S_;
    float* proj      = summed + (size_t)B_ * S_ * H_;

    const int NBIAS = B_ * T_ * S_;
    zero_f32_kernel<<<(NBIAS + 255) / 256, 256, 0, stream>>>(bias_flat, NBIAS);
    bias_proj_kernel<<<1, 64, 0, stream>>>(bias_embs, bias_scalar, proj);
    edge_scatter_kernel<<<(E + 255) / 256, 256, 0, stream>>>(edges0, edges1, proj,
                                                             bias_flat, E);
    summed_keys_kernel<<<(B_ * S_ * H_ + 255) / 256, 256, 0, stream>>>(key, summed);

    scores_wmma_kernel<<<dim3(S_ / 64, T_ / 16, B_ * H_), 128, 0, stream>>>(
        query, key, mask, bias_flat, summed, alpha);

    softmax_rows_kernel<<<B_ * H_ * T_, 256, 0, stream>>>(alpha);

    pv_wmma_kernel<<<dim3(T_ / 16, B_ * H_), 128, 0, stream>>>(
        value, alpha, out_f);
}